// JacobiPool_26121991094891
// MI455X (gfx1250) — compile-verified
//
#include <hip/hip_runtime.h>
#include <math.h>

// ---------------------------------------------------------------------------
// JacobiPool for gfx1250 (MI455X).
// N=100000 nodes, C=128 feats, E=3.2M edges, K=3, RATIO=0.8.
// Fused projection x@[W|lin_w] uses V_WMMA_F32_16X16X4_F32 (f32 WMMA) with a
// branchless inner loop (LDS-staged B, clamped A rows).
// Edge passes use f32 global atomics; exact ordered top-k via bitonic sort.
// ---------------------------------------------------------------------------

typedef float v2f __attribute__((ext_vector_type(2)));
typedef float v8f __attribute__((ext_vector_type(8)));

#define NEG_SLOPE 0.2f
#define CFEAT 128

// ---- float <-> order-preserving uint (for atomic float max) ----------------
__device__ __forceinline__ unsigned f2ord(float f) {
    unsigned u = __float_as_uint(f);
    return (u & 0x80000000u) ? ~u : (u | 0x80000000u);
}
__device__ __forceinline__ float ord2f(unsigned u) {
    unsigned v = (u & 0x80000000u) ? (u & 0x7FFFFFFFu) : ~u;
    return __uint_as_float(v);
}

// ---- init all node-sized accumulators --------------------------------------
__global__ void init_kernel(unsigned* __restrict__ m_u, float* __restrict__ s,
                            float* __restrict__ deg, float* __restrict__ agg,
                            int* __restrict__ inv, int n) {
    int i = blockIdx.x * blockDim.x + threadIdx.x;
    if (i >= n) return;
    m_u[i] = 0u;          // below ordered-encoding of any finite float
    s[i]   = 0.f;
    deg[i] = 0.f;
    agg[i] = 0.f;
    inv[i] = -1;
}

// ---- fused projection: xp = x@W, x_hat = x@lin_w + b, via f32 WMMA ---------
// Each wave computes a 16-row tile. A (16x4 f32) per ISA layout:
//   VGPR0: lanes0-15 -> (M=lane, K=0), lanes16-31 -> (M=lane-16, K=2)
//   VGPR1: K=1 / K=3.  B (4x16) mirrored: VGPR0 K=2*half, VGPR1 K=2*half+1,
//   N = lane&15.  Column 0 = W, column 1 = lin_w, columns 2..15 = 0.
// Inner loop is branchless: B built from LDS with 0/1 select masks, A rows
// clamped (garbage rows never stored).
__global__ __launch_bounds__(128)
void proj_wmma_kernel(const float* __restrict__ x,
                      const float* __restrict__ W,
                      const float* __restrict__ lin_w,
                      const float* __restrict__ att_src_p,
                      const float* __restrict__ att_dst_p,
                      const float* __restrict__ lin_b_p,
                      float* __restrict__ a_s, float* __restrict__ a_d,
                      float* __restrict__ x_hat, int n)
{
    __shared__ float sW[CFEAT];
    __shared__ float sL[CFEAT];
    if (threadIdx.x < CFEAT) {
        sW[threadIdx.x] = W[threadIdx.x];
        sL[threadIdx.x] = lin_w[threadIdx.x];
    }
    __syncthreads();

    const int wave = threadIdx.x >> 5;              // 4 waves / block
    const int lane = threadIdx.x & 31;
    const int lr   = lane & 15;
    const int hf   = lane >> 4;                     // 0: K=0,1  1: K=2,3
    const int rowBase = (blockIdx.x * 4 + wave) * 16;
    const int arow = min(rowBase + lr, n - 1);      // clamped A row (stores guarded)
    const float* __restrict__ xrow = x + (size_t)arow * CFEAT + hf * 2;

    const float selW = (lr == 0) ? 1.f : 0.f;       // B column-select masks
    const float selL = (lr == 1) ? 1.f : 0.f;

    v8f acc = {};
    #pragma unroll 4
    for (int kk = 0; kk < CFEAT; kk += 4) {
        const int ka = kk + hf * 2;
        const float2 av = *(const float2*)(xrow + kk);   // 8B-aligned (ka even)
        v2f a; a.x = av.x; a.y = av.y;
        v2f b;
        b.x = selW * sW[ka]     + selL * sL[ka];
        b.y = selW * sW[ka + 1] + selL * sL[ka + 1];
        acc = __builtin_amdgcn_wmma_f32_16x16x4_f32(
                  false, a, false, b, (short)0, acc, false, false);
    }

    // C/D layout: VGPR v, lanes0-15 -> (M=v, N=lane); lanes16-31 -> (M=v+8, N=lane-16)
    const float asc = *att_src_p, adc = *att_dst_p, lb = *lin_b_p;
    if (lr < 2) {
        #pragma unroll
        for (int v = 0; v < 8; ++v) {
            const int r = rowBase + hf * 8 + v;
            if (r < n) {
                const float d = acc[v];
                if (lr == 0) { a_s[r] = d * asc; a_d[r] = d * adc; }
                else         { x_hat[r] = d + lb; }
            }
        }
    }
}

// ---- edge pass 1: e = leaky_relu(a_s[src] + a_d[dst]); segment max over dst
__global__ void edge_max_kernel(const int* __restrict__ ei,
                                const float* __restrict__ a_s,
                                const float* __restrict__ a_d,
                                float* __restrict__ e_arr,
                                unsigned* __restrict__ m_u, int E, int n) {
    int e = blockIdx.x * blockDim.x + threadIdx.x;
    if (e >= E + n) return;
    const int src = (e < E) ? ei[e]     : (e - E);
    const int dst = (e < E) ? ei[E + e] : (e - E);
    float v = a_s[src] + a_d[dst];
    v = (v > 0.f) ? v : NEG_SLOPE * v;
    e_arr[e] = v;
    atomicMax(&m_u[dst], f2ord(v));
}

// ---- edge pass 2: ex = exp(e - m[dst]); segment sum over dst ---------------
__global__ void edge_expsum_kernel(const int* __restrict__ ei,
                                   const unsigned* __restrict__ m_u,
                                   float* __restrict__ e_arr,
                                   float* __restrict__ s, int E, int n) {
    int e = blockIdx.x * blockDim.x + threadIdx.x;
    if (e >= E + n) return;
    const int dst = (e < E) ? ei[E + e] : (e - E);
    const float ex = expf(e_arr[e] - ord2f(m_u[dst]));
    e_arr[e] = ex;
    atomicAdd(&s[dst], ex);
}

// ---- edge pass 3: att = ex / s[dst]; deg[src] += att -----------------------
__global__ void edge_att_deg_kernel(const int* __restrict__ ei,
                                    const float* __restrict__ s,
                                    float* __restrict__ e_arr,
                                    float* __restrict__ deg, int E, int n) {
    int e = blockIdx.x * blockDim.x + threadIdx.x;
    if (e >= E + n) return;
    const int src = (e < E) ? ei[e]     : (e - E);
    const int dst = (e < E) ? ei[E + e] : (e - E);
    const float att = e_arr[e] / s[dst];
    e_arr[e] = att;
    atomicAdd(&deg[src], att);
}

// ---- dinv = rsqrt(deg) in place --------------------------------------------
__global__ void node_dinv_kernel(float* __restrict__ deg, int n) {
    int i = blockIdx.x * blockDim.x + threadIdx.x;
    if (i < n) deg[i] = rsqrtf(deg[i]);
}

// ---- edge pass 4: Chebyshev poly on A = -dinv[s]*att*dinv[d]; agg scatter --
__global__ void edge_poly_kernel(const int* __restrict__ ei,
                                 const float* __restrict__ dinv,
                                 const float* __restrict__ e_arr,
                                 const float* __restrict__ x_hat,
                                 const float* __restrict__ alphas,
                                 float* __restrict__ agg, int E, int n) {
    int e = blockIdx.x * blockDim.x + threadIdx.x;
    if (e >= E + n) return;
    const int src = (e < E) ? ei[e]     : (e - E);
    const int dst = (e < E) ? ei[E + e] : (e - E);
    const float A = -(dinv[src] * e_arr[e] * dinv[dst]);
    float tprev = (src == dst) ? 1.f : 0.f;
    float tcur  = A;
    float poly  = alphas[0] * tprev + alphas[1] * tcur;
    float tnext = 2.f * A * tcur - tprev;          // k = 2
    poly += alphas[2] * tnext; tprev = tcur; tcur = tnext;
    tnext = 2.f * A * tcur - tprev;                // k = 3
    poly += alphas[3] * tnext;
    atomicAdd(&agg[src], poly * x_hat[dst]);
}

// ---- pack (descending key, ascending index) into 64-bit sort items ---------
__global__ void pack_kernel(const float* __restrict__ agg,
                            unsigned long long* __restrict__ items, int n, int M) {
    int i = blockIdx.x * blockDim.x + threadIdx.x;
    if (i >= M) return;
    const unsigned key = (i < n) ? ~f2ord(agg[i]) : 0xFFFFFFFFu; // padding sorts last
    items[i] = ((unsigned long long)key << 32) | (unsigned)i;
}

// ---- bitonic sort: global step (j >= 2048) ---------------------------------
__global__ void bitonic_global_kernel(unsigned long long* __restrict__ items,
                                      int k, int j, int M) {
    int h = blockIdx.x * blockDim.x + threadIdx.x;
    if (h >= (M >> 1)) return;
    const int i = ((h & ~(j - 1)) << 1) | (h & (j - 1));
    const int p = i + j;
    const bool asc = ((i & k) == 0);
    const unsigned long long a = items[i], b = items[p];
    if (asc == (a > b)) { items[i] = b; items[p] = a; }
}

// ---- bitonic sort: LDS-staged steps (jstart <= 1024 down to 1) -------------
__global__ __launch_bounds__(1024)
void bitonic_lds_kernel(unsigned long long* __restrict__ items,
                        int k, int jstart, int M) {
    __shared__ unsigned long long sh[2048];
    const int base = blockIdx.x * 2048;
    sh[threadIdx.x]        = items[base + threadIdx.x];
    sh[threadIdx.x + 1024] = items[base + threadIdx.x + 1024];
    __syncthreads();
    for (int j = jstart; j >= 1; j >>= 1) {
        const int li = ((threadIdx.x & ~(j - 1)) << 1) | (threadIdx.x & (j - 1));
        const bool asc = (((base + li) & k) == 0);
        const unsigned long long a = sh[li], b = sh[li + j];
        if (asc == (a > b)) { sh[li] = b; sh[li + j] = a; }
        __syncthreads();
    }
    items[base + threadIdx.x]        = sh[threadIdx.x];
    items[base + threadIdx.x + 1024] = sh[threadIdx.x + 1024];
}

// ---- perm / inv / batch ----------------------------------------------------
__global__ void perm_kernel(const unsigned long long* __restrict__ items,
                            int* __restrict__ perm_out, int* __restrict__ batch_out,
                            int* __restrict__ inv, int k_keep) {
    int i = blockIdx.x * blockDim.x + threadIdx.x;
    if (i >= k_keep) return;
    const int idx = (int)(items[i] & 0xFFFFFFFFull);
    perm_out[i]  = idx;
    batch_out[i] = 0;
    inv[idx]     = i;
}

// ---- x_out = x[perm] * tanh(agg[perm]) -------------------------------------
__global__ void xout_kernel(const float* __restrict__ x,
                            const float* __restrict__ agg,
                            const int* __restrict__ perm,
                            float* __restrict__ x_out, int k_keep) {
    int t = blockIdx.x * blockDim.x + threadIdx.x;
    if (t >= k_keep * CFEAT) return;
    const int i = t >> 7, c = t & (CFEAT - 1);
    const int row = perm[i];
    x_out[t] = x[(size_t)row * CFEAT + c] * tanhf(agg[row]);
}

// ---- filter_adj: relabel kept edges, -1 elsewhere --------------------------
__global__ void filter_kernel(const int* __restrict__ ei,
                              const int* __restrict__ inv,
                              int* __restrict__ out_ei, int E) {
    int e = blockIdx.x * blockDim.x + threadIdx.x;
    if (e >= E) return;
    const int ns = inv[ei[e]], nd = inv[ei[E + e]];
    const bool ok = (ns >= 0) && (nd >= 0);
    out_ei[e]     = ok ? ns : -1;
    out_ei[E + e] = ok ? nd : -1;
}

// ---------------------------------------------------------------------------
extern "C" void kernel_launch(void* const* d_in, const int* in_sizes, int n_in,
                              void* d_out, int out_size, void* d_ws, size_t ws_size,
                              hipStream_t stream) {
    const float* x       = (const float*)d_in[0];
    const int*   ei      = (const int*)  d_in[1];
    const float* W       = (const float*)d_in[2];
    const float* att_src = (const float*)d_in[3];
    const float* att_dst = (const float*)d_in[4];
    const float* lin_w   = (const float*)d_in[5];
    const float* lin_b   = (const float*)d_in[6];
    const float* alphas  = (const float*)d_in[7];

    const int n = in_sizes[0] / CFEAT;      // 100000
    const int E = in_sizes[1] / 2;          // 3200000
    const int k_keep = (int)((4LL * n + 4) / 5);   // ceil(0.8*n)
    int M = 1; while (M < n) M <<= 1;       // sort size (pow2), 131072

    // workspace layout (items first for 8-byte alignment)
    char* ws = (char*)d_ws;
    unsigned long long* items = (unsigned long long*)ws;
    size_t off = (size_t)M * 8;
    float*    a_s   = (float*)   (ws + off); off += (size_t)n * 4;
    float*    a_d   = (float*)   (ws + off); off += (size_t)n * 4;
    float*    x_hat = (float*)   (ws + off); off += (size_t)n * 4;
    unsigned* m_u   = (unsigned*)(ws + off); off += (size_t)n * 4;
    float*    s     = (float*)   (ws + off); off += (size_t)n * 4;
    float*    deg   = (float*)   (ws + off); off += (size_t)n * 4;  // becomes dinv
    float*    agg   = (float*)   (ws + off); off += (size_t)n * 4;
    int*      inv   = (int*)     (ws + off); off += (size_t)n * 4;
    float*    e_arr = (float*)   (ws + off); off += (size_t)(E + n) * 4;

    // output layout: [x_out f32 | new_ei i32 | batch i32 | perm i32]
    float* x_out     = (float*)d_out;
    int*   out_ei    = (int*)d_out + (size_t)k_keep * CFEAT;
    int*   batch_out = out_ei + (size_t)2 * E;
    int*   perm_out  = batch_out + k_keep;

    const int EN = E + n;
    const int TB = 256;

    init_kernel<<<(n + TB - 1) / TB, TB, 0, stream>>>(m_u, s, deg, agg, inv, n);
    proj_wmma_kernel<<<(n + 63) / 64, 128, 0, stream>>>(
        x, W, lin_w, att_src, att_dst, lin_b, a_s, a_d, x_hat, n);
    edge_max_kernel<<<(EN + TB - 1) / TB, TB, 0, stream>>>(ei, a_s, a_d, e_arr, m_u, E, n);
    edge_expsum_kernel<<<(EN + TB - 1) / TB, TB, 0, stream>>>(ei, m_u, e_arr, s, E, n);
    edge_att_deg_kernel<<<(EN + TB - 1) / TB, TB, 0, stream>>>(ei, s, e_arr, deg, E, n);
    node_dinv_kernel<<<(n + TB - 1) / TB, TB, 0, stream>>>(deg, n);
    edge_poly_kernel<<<(EN + TB - 1) / TB, TB, 0, stream>>>(
        ei, deg, e_arr, x_hat, alphas, agg, E, n);

    pack_kernel<<<(M + TB - 1) / TB, TB, 0, stream>>>(agg, items, n, M);
    for (int k = 2; k <= M; k <<= 1) {
        int j = k >> 1;
        for (; j >= 2048; j >>= 1)
            bitonic_global_kernel<<<((M >> 1) + TB - 1) / TB, TB, 0, stream>>>(items, k, j, M);
        bitonic_lds_kernel<<<M / 2048, 1024, 0, stream>>>(items, k, j, M);
    }

    perm_kernel<<<(k_keep + TB - 1) / TB, TB, 0, stream>>>(items, perm_out, batch_out, inv, k_keep);
    xout_kernel<<<((size_t)k_keep * CFEAT + TB - 1) / TB, TB, 0, stream>>>(
        x, agg, perm_out, x_out, k_keep);
    filter_kernel<<<(E + TB - 1) / TB, TB, 0, stream>>>(ei, inv, out_ei, E);
}